// Transe_42803644072143
// MI455X (gfx1250) — compile-verified
//
#include <hip/hip_runtime.h>
#include <math.h>

typedef __attribute__((ext_vector_type(2))) float v2f;
typedef __attribute__((ext_vector_type(8))) float v8f;

#define MARGIN 9.0f
#define D 64
#define BROWS 32

// ---------------------------------------------------------------------------
// Kernel 1: q = s + r  (32 x 64), qn[b] = ||q[b]||^2
// 1 block of 1024 threads = 32 waves (wave32); wave w owns row w.
// ---------------------------------------------------------------------------
__global__ __launch_bounds__(1024) void transe_prep(
    const float* __restrict__ s, const float* __restrict__ r,
    float* __restrict__ q, float* __restrict__ qn) {
  int w = threadIdx.x >> 5;
  int lane = threadIdx.x & 31;
  int base = w * D + 2 * lane;
  float x0 = s[base] + r[base];
  float x1 = s[base + 1] + r[base + 1];
  q[base] = x0;
  q[base + 1] = x1;
  float p = x0 * x0 + x1 * x1;
  #pragma unroll
  for (int off = 16; off >= 1; off >>= 1)
    p += __shfl_xor(p, off, 32);
  if (lane == 0) qn[w] = p;
}

// ---------------------------------------------------------------------------
// Kernel 2: pred[b,n] = MARGIN - sqrt(qn[b] + ||e[n]||^2 - 2 * q[b].e[n])
// Block = 256 threads (8 waves). Wave computes a 32x16 output tile via
// two V_WMMA_F32_16X16X4_F32 accumulators over a K=64 loop (16 steps).
// emb_e is read exactly once; ||e||^2 folded into the B-fragment loads.
// ---------------------------------------------------------------------------
__global__ __launch_bounds__(256) void transe_score(
    const float* __restrict__ e, const float* __restrict__ q,
    const float* __restrict__ qn, float* __restrict__ out, int N) {
  __shared__ __attribute__((aligned(16))) float s_q[BROWS * D];  // 8 KB
  __shared__ float s_qn[BROWS];

  int tid = threadIdx.x;

  // Stage q into LDS: 2048 floats via 2x float4 per thread.
  {
    const float4* src = (const float4*)q;
    float4* dst = (float4*)s_q;
    dst[tid] = src[tid];
    dst[tid + 256] = src[tid + 256];
  }
  if (tid < BROWS) s_qn[tid] = qn[tid];
  __syncthreads();

  int wave = tid >> 5;
  int lane = tid & 31;
  int col = lane & 15;       // N within 16-wide tile / M within A frag
  int hi = lane >> 4;        // K-half selector (K offsets {0,1} vs {2,3})

  long n0 = (long)blockIdx.x * 128 + (long)wave * 16;
  long erow = n0 + col;
  if (erow > (long)N - 1) erow = (long)N - 1;  // clamp: keep EXEC all-ones

  const float* ebase = e + erow * D + hi * 2;        // B fragment source
  const float* abase0 = s_q + col * D + hi * 2;      // A frag, rows 0..15
  const float* abase1 = abase0 + 16 * D;             // A frag, rows 16..31

  v8f acc0 = {};
  v8f acc1 = {};
  float en = 0.0f;  // partial ||e[col]||^2 (this lane's K-half)

  #pragma unroll
  for (int k = 0; k < D; k += 4) {
    v2f b = *(const v2f*)(ebase + k);
    en = fmaf(b.x, b.x, en);
    en = fmaf(b.y, b.y, en);
    v2f a0 = *(const v2f*)(abase0 + k);
    v2f a1 = *(const v2f*)(abase1 + k);
    acc0 = __builtin_amdgcn_wmma_f32_16x16x4_f32(
        false, a0, false, b, (short)0, acc0, false, false);
    acc1 = __builtin_amdgcn_wmma_f32_16x16x4_f32(
        false, a1, false, b, (short)0, acc1, false, false);
  }

  // Combine the two K-halves: lanes L and L^16 hold complementary partials.
  en += __shfl_xor(en, 16, 32);

  long n = n0 + col;
  if (n < (long)N) {
    #pragma unroll
    for (int i = 0; i < 8; ++i) {
      int m0 = i + hi * 8;  // accumulator row mapping (C/D layout)
      float d0 = s_qn[m0] + en - 2.0f * acc0[i];
      d0 = d0 > 0.0f ? d0 : 0.0f;
      out[(long)m0 * N + n] = MARGIN - sqrtf(d0);

      int m1 = m0 + 16;
      float d1 = s_qn[m1] + en - 2.0f * acc1[i];
      d1 = d1 > 0.0f ? d1 : 0.0f;
      out[(long)m1 * N + n] = MARGIN - sqrtf(d1);
    }
  }
}

// ---------------------------------------------------------------------------
extern "C" void kernel_launch(void* const* d_in, const int* in_sizes, int n_in,
                              void* d_out, int out_size, void* d_ws, size_t ws_size,
                              hipStream_t stream) {
  const float* s_emb = (const float*)d_in[0];
  const float* rel_emb = (const float*)d_in[1];
  const float* emb_e = (const float*)d_in[2];
  float* out = (float*)d_out;

  int N = in_sizes[2] / D;  // 200000

  float* q = (float*)d_ws;            // 32*64 floats
  float* qn = q + BROWS * D;          // 32 floats

  transe_prep<<<1, 1024, 0, stream>>>(s_emb, rel_emb, q, qn);

  int grid = (N + 127) / 128;
  transe_score<<<grid, 256, 0, stream>>>(emb_e, q, qn, out, N);
}